// FlexAttention_9878424781143
// MI455X (gfx1250) — compile-verified
//
#include <hip/hip_runtime.h>

// Sliding-window causal attention (flash-style) for MI455X / gfx1250.
// B=2, H=16, S=2048, D=64, WINDOW=512, fp32 in/out.
//  - f32 WMMA 16x16x4 for QK^T, P@V, and P@ones (softmax denominator)
//  - K/V tiles: double-buffered GLOBAL_LOAD_ASYNC_TO_LDS (ASYNCcnt-tracked)
//  - masking only on the two edge tiles of each wave's window

#define BB 2
#define HH 16
#define SS 2048
#define DD 64
#define WW 512

typedef __attribute__((ext_vector_type(2))) float v2f;
typedef __attribute__((ext_vector_type(8))) float v8f;
typedef __attribute__((__vector_size__(4 * sizeof(int)))) int v4i_t;

#define LOG2E 1.4426950408889634f

#if defined(__has_builtin)
#if __has_builtin(__builtin_amdgcn_global_load_async_to_lds_b128)
#define HAVE_ASYNC_LDS 1
#endif
#endif
#ifndef HAVE_ASYNC_LDS
#define HAVE_ASYNC_LDS 0
#endif

__device__ __forceinline__ int imax(int a, int b) { return a > b ? a : b; }

#if HAVE_ASYNC_LDS
__device__ __forceinline__ void async_cp16(const float* g, float* l) {
    // one lane copies 16B global -> LDS, tracked by ASYNCcnt
    __builtin_amdgcn_global_load_async_to_lds_b128(
        (__attribute__((address_space(1))) v4i_t*)(g),
        (__attribute__((address_space(3))) v4i_t*)(l), 0, 0);
}
#endif

__global__ __launch_bounds__(128) void swa_fwd_kernel(
    const float* __restrict__ Q, const float* __restrict__ K,
    const float* __restrict__ V, float* __restrict__ O)
{
    // double-buffered K/V tiles (16 rows x 64 cols, stride 68 to kill bank conflicts)
    __shared__ float kbuf[2][16][68];
    __shared__ float vbuf[2][16][68];
    // per-wave P transpose scratch (16x16, stride 18: even for v2f align, conflict-free)
    __shared__ float pstage[4][16][18];

    const int tid  = threadIdx.x;
    const int wave = tid >> 5;
    const int lane = tid & 31;
    const int n    = lane & 15;   // column index within 16 (and A-matrix row)
    const int hi   = lane >> 4;   // half-wave select

    const int bh        = blockIdx.x >> 5;   // 0..31  (b*H + h)
    const int qblk      = blockIdx.x & 31;   // 0..31  (64-query block)
    const int qbase_blk = qblk * 64;
    const int qbase     = qbase_blk + wave * 16;

    const float* Qb = Q + (size_t)bh * SS * DD;
    const float* Kb = K + (size_t)bh * SS * DD;
    const float* Vb = V + (size_t)bh * SS * DD;
    float*       Ob = O + (size_t)bh * SS * DD;

    // ---- Q fragments in 16x16x4 f32 WMMA A-layout ----
    // pre-scaled by (1/sqrt(D)) * log2(e) so softmax runs natively in exp2 domain
    v2f qa[16];
    {
        const float* qrow = Qb + (qbase + n) * DD + 2 * hi;
        const float qscale = 0.125f * LOG2E;
        #pragma unroll
        for (int kk = 0; kk < 16; ++kk) {
            v2f q = *(const v2f*)(qrow + kk * 4);
            qa[kk] = q * qscale;
        }
    }

    // key-tile ranges: block union and this wave's window
    const int t_lo  = imax(0, qblk * 4 - (WW / 16));
    const int t_hi  = qblk * 4 + 3;
    const int aa    = qbase >> 4;                 // this wave's causal tile index
    const int wt_lo = imax(0, aa - (WW / 16));
    const int wt_hi = aa;

    // accumulators in WMMA C-layout; row = v + 8*hi, col = nb*16+n
    // o4 = P @ ones accumulates the softmax denominator (every column = row sum)
    v8f o0 = {}, o1 = {}, o2 = {}, o3 = {}, o4 = {};
    float mstate[8];
    #pragma unroll
    for (int v = 0; v < 8; ++v) mstate[v] = -1e30f;

    // tile staging: each of 128 threads owns 8 floats (32B) of each 16x64 tile
    const int sr = tid >> 3;        // 0..15 tile row
    const int sc = (tid & 7) * 8;   // 0..56 tile col

#if HAVE_ASYNC_LDS
    auto issue_tile = [&](int t, int buf) {
        const float* kp = Kb + (t * 16 + sr) * DD + sc;
        const float* vp = Vb + (t * 16 + sr) * DD + sc;
        async_cp16(kp,     &kbuf[buf][sr][sc]);
        async_cp16(kp + 4, &kbuf[buf][sr][sc + 4]);
        async_cp16(vp,     &vbuf[buf][sr][sc]);
        async_cp16(vp + 4, &vbuf[buf][sr][sc + 4]);
    };
#else
    float4 kr0, kr1, vr0, vr1;
    auto g2r = [&](int t) {
        const float* kp = Kb + (t * 16 + sr) * DD + sc;
        const float* vp = Vb + (t * 16 + sr) * DD + sc;
        kr0 = *(const float4*)(kp);
        kr1 = *(const float4*)(kp + 4);
        vr0 = *(const float4*)(vp);
        vr1 = *(const float4*)(vp + 4);
    };
    auto r2s = [&](int buf) {
        *(float4*)&kbuf[buf][sr][sc]     = kr0;
        *(float4*)&kbuf[buf][sr][sc + 4] = kr1;
        *(float4*)&vbuf[buf][sr][sc]     = vr0;
        *(float4*)&vbuf[buf][sr][sc + 4] = vr1;
    };
#endif

    int buf = 0;
#if HAVE_ASYNC_LDS
    issue_tile(t_lo, 0);
#else
    g2r(t_lo);
#endif

    for (int t = t_lo; t <= t_hi; ++t) {
#if HAVE_ASYNC_LDS
        // my slice of tile t landed in LDS; then barrier makes all slices visible
        asm volatile("s_wait_asynccnt 0" ::: "memory");
        __syncthreads();
        if (t < t_hi) issue_tile(t + 1, buf ^ 1);  // overlap copy(t+1) with compute(t)
#else
        r2s(buf);
        if (t < t_hi) g2r(t + 1);
        __syncthreads();
#endif

        if (t >= wt_lo && t <= wt_hi) {   // wave-uniform: EXEC stays all-ones
            const int kbase = t * 16;

            // ---------------- S2 = (Q*scale*log2e) @ K^T : f32 WMMA ----------------
            v8f acc = {};
            #pragma unroll
            for (int kk = 0; kk < 16; ++kk) {
                // B-frag: lane n holds K[kbase+n][kk*4 + 2*hi + {0,1}]
                v2f b = *(const v2f*)&kbuf[buf][n][kk * 4 + 2 * hi];
                acc = __builtin_amdgcn_wmma_f32_16x16x4_f32(
                        false, qa[kk], false, b, (short)0, acc, false, false);
            }

            // ---- mask: only the causal tile (t==aa) and the window-tail tile ----
            float s[8];
            if (t == aa || t == aa - (WW / 16)) {          // wave-uniform edge test
                const int d0 = qbase + 8 * hi - kbase - n; // (qi - kj) - v
                #pragma unroll
                for (int v = 0; v < 8; ++v)
                    s[v] = ((unsigned)(d0 + v) < WW) ? acc[v] : -INFINITY;
            } else {
                #pragma unroll
                for (int v = 0; v < 8; ++v) s[v] = acc[v];
            }

            // ---------------- online softmax (exp2 domain) ----------------
            float p[8], alpha[8];
            #pragma unroll
            for (int v = 0; v < 8; ++v) {
                // row-max over the 16 lanes of this half-wave
                float rmax = s[v];
                rmax = fmaxf(rmax, __shfl_xor(rmax, 1));
                rmax = fmaxf(rmax, __shfl_xor(rmax, 2));
                rmax = fmaxf(rmax, __shfl_xor(rmax, 4));
                rmax = fmaxf(rmax, __shfl_xor(rmax, 8));
                const float mnew = fmaxf(mstate[v], rmax);
                alpha[v] = exp2f(mstate[v] - mnew);
                p[v]     = exp2f(s[v] - mnew);
                mstate[v] = mnew;
            }
            // rescale running O (and denominator) by alpha per-row
            #pragma unroll
            for (int v = 0; v < 8; ++v) {
                o0[v] *= alpha[v]; o1[v] *= alpha[v];
                o2[v] *= alpha[v]; o3[v] *= alpha[v];
                o4[v] *= alpha[v];
            }

            // ---- P: C-layout -> A-layout via per-wave LDS scratch ----
            #pragma unroll
            for (int v = 0; v < 8; ++v)
                pstage[wave][v + 8 * hi][n] = p[v];
            asm volatile("s_wait_dscnt 0" ::: "memory");  // wave-local LDS fence
            v2f pa[4];
            #pragma unroll
            for (int kk = 0; kk < 4; ++kk)
                pa[kk] = *(const v2f*)&pstage[wave][n][kk * 4 + 2 * hi];

            // ------- O += P @ V ; denominator += P @ ones : f32 WMMA -------
            const v2f onef = {1.0f, 1.0f};
            #pragma unroll
            for (int kk = 0; kk < 4; ++kk) {
                const int r0 = kk * 4 + 2 * hi;
                v2f b0, b1, b2, b3;
                b0.x = vbuf[buf][r0][n];          b0.y = vbuf[buf][r0 + 1][n];
                b1.x = vbuf[buf][r0][16 + n];     b1.y = vbuf[buf][r0 + 1][16 + n];
                b2.x = vbuf[buf][r0][32 + n];     b2.y = vbuf[buf][r0 + 1][32 + n];
                b3.x = vbuf[buf][r0][48 + n];     b3.y = vbuf[buf][r0 + 1][48 + n];
                o0 = __builtin_amdgcn_wmma_f32_16x16x4_f32(false, pa[kk], false, b0,   (short)0, o0, false, false);
                o1 = __builtin_amdgcn_wmma_f32_16x16x4_f32(false, pa[kk], false, b1,   (short)0, o1, false, false);
                o2 = __builtin_amdgcn_wmma_f32_16x16x4_f32(false, pa[kk], false, b2,   (short)0, o2, false, false);
                o3 = __builtin_amdgcn_wmma_f32_16x16x4_f32(false, pa[kk], false, b3,   (short)0, o3, false, false);
                o4 = __builtin_amdgcn_wmma_f32_16x16x4_f32(false, pa[kk], false, onef, (short)0, o4, false, false);
            }
        }
        buf ^= 1;
    }

    // ---------------- epilogue: O / denom, coalesced stores ----------------
    #pragma unroll
    for (int v = 0; v < 8; ++v) {
        const float inv = 1.0f / o4[v];
        float* orow = Ob + (qbase + v + 8 * hi) * DD;
        orow[n]      = o0[v] * inv;
        orow[16 + n] = o1[v] * inv;
        orow[32 + n] = o2[v] * inv;
        orow[48 + n] = o3[v] * inv;
    }
}

extern "C" void kernel_launch(void* const* d_in, const int* in_sizes, int n_in,
                              void* d_out, int out_size, void* d_ws, size_t ws_size,
                              hipStream_t stream) {
    const float* Q = (const float*)d_in[0];
    const float* K = (const float*)d_in[1];
    const float* V = (const float*)d_in[2];
    float*       O = (float*)d_out;
    (void)in_sizes; (void)n_in; (void)out_size; (void)d_ws; (void)ws_size;

    dim3 grid(BB * HH * (SS / 64));  // 1024 blocks: one 64-query block each
    dim3 block(128);                 // 4 waves; one 16-query tile per wave
    swa_fwd_kernel<<<grid, block, 0, stream>>>(Q, K, V, O);
}